// HeteroGNN_18245021074001
// MI455X (gfx1250) — compile-verified
//
#include <hip/hip_runtime.h>
#include <stddef.h>

// ---------------- constants from the reference ----------------
#define N_GENE 100000
#define N_DIS  50000
#define F_IN   128
#define H_DIM  64
#define E_GDA  2000000
#define E_PPI  3000000
#define E_LBL  500000
#define BN_EPS 1e-5f
#define SLOPE  0.01f

typedef float v2f __attribute__((ext_vector_type(2)));
typedef float v8f __attribute__((ext_vector_type(8)));

// ---------------- fused weight precompute --------------------
// A[k,j] = sum_h Wd[k,h]*Wu[h,j]        (k < Fdst)
// B[k,j] = sum_h Ws[k,h]*Wu[64+h,j]     (k < Fsrc)
// c[j]   = sum_h bd[h]*Wu[h,j] + bs[h]*Wu[64+h,j] + bu[j]
__global__ void fuse_weights_kernel(const float* __restrict__ Ws, const float* __restrict__ bs,
                                    const float* __restrict__ Wd, const float* __restrict__ bd,
                                    const float* __restrict__ Wu, const float* __restrict__ bu,
                                    float* __restrict__ Af, float* __restrict__ Bf,
                                    float* __restrict__ cf, int Fdst, int Fsrc) {
    int tid = blockIdx.x * blockDim.x + threadIdx.x;
    int Fmax = Fdst > Fsrc ? Fdst : Fsrc;
    if (tid >= Fmax * 64) return;
    int k = tid >> 6, j = tid & 63;
    if (k < Fdst) {
        float s = 0.f;
        for (int h = 0; h < 64; ++h) s += Wd[k * 64 + h] * Wu[h * 64 + j];
        Af[k * 64 + j] = s;
    }
    if (k < Fsrc) {
        float s = 0.f;
        for (int h = 0; h < 64; ++h) s += Ws[k * 64 + h] * Wu[(64 + h) * 64 + j];
        Bf[k * 64 + j] = s;
    }
    if (k == 0) {
        float s = bu[j];
        for (int h = 0; h < 64; ++h)
            s += bd[h] * Wu[h * 64 + j] + bs[h] * Wu[(64 + h) * 64 + j];
        cf[j] = s;
    }
}

// ---------------- edge scatter-add (mean numerator + count) ----
__global__ void scatter_add_kernel(const float* __restrict__ xsrc, const int* __restrict__ ei,
                                   float* __restrict__ aggr, float* __restrict__ cnt,
                                   int E, int F) {
    long long tid = blockIdx.x * (long long)blockDim.x + threadIdx.x;
    int cpe = F >> 2;  // float4 chunks per edge
    long long total = (long long)E * cpe;
    if (tid >= total) return;
    int e = (int)(tid / cpe);
    int c = (int)(tid % cpe);
    int src = ei[e];
    int dst = ei[E + e];
    const float4 v = *(const float4*)(xsrc + (size_t)src * F + c * 4);
    float* ap = aggr + (size_t)dst * F + c * 4;
    atomicAdd(ap + 0, v.x);
    atomicAdd(ap + 1, v.y);
    atomicAdd(ap + 2, v.z);
    atomicAdd(ap + 3, v.w);
    if (c == 0) atomicAdd(cnt + dst, 1.0f);
}

__global__ void mean_div_kernel(float* __restrict__ aggr, const float* __restrict__ cnt,
                                long long total, int logF) {
    long long i = blockIdx.x * (long long)blockDim.x + threadIdx.x;
    if (i >= total) return;
    float c = cnt[i >> logF];
    aggr[i] = (c > 0.f) ? aggr[i] / c : 0.f;
}

// ---------------- WMMA conv GEMM -----------------------------
// out[N,64] = Xd[N,K]@A + Ag[N,K]@B + bias   (combine=1: out=(old+res)*0.5)
__device__ inline void wmma_accum(const float* __restrict__ X, const float* __restrict__ W,
                                  int rowA, int K, int m, int khi,
                                  v8f& c0, v8f& c1, v8f& c2, v8f& c3) {
    const float* xrow = X + (size_t)rowA * K;
    for (int k0 = 0; k0 < K; k0 += 4) {
        // A frag (16x4 f32): lanes 0-15 hold K=k0,k0+1 ; lanes 16-31 hold K=k0+2,k0+3
        v2f a;
        a.x = xrow[k0 + 2 * khi];
        a.y = xrow[k0 + 2 * khi + 1];
        const float* bp = W + (size_t)(k0 + 2 * khi) * 64 + m;
        v2f b0; b0.x = bp[0];  b0.y = bp[64];
        v2f b1; b1.x = bp[16]; b1.y = bp[80];
        v2f b2; b2.x = bp[32]; b2.y = bp[96];
        v2f b3; b3.x = bp[48]; b3.y = bp[112];
        c0 = __builtin_amdgcn_wmma_f32_16x16x4_f32(false, a, false, b0, (short)0, c0, false, false);
        c1 = __builtin_amdgcn_wmma_f32_16x16x4_f32(false, a, false, b1, (short)0, c1, false, false);
        c2 = __builtin_amdgcn_wmma_f32_16x16x4_f32(false, a, false, b2, (short)0, c2, false, false);
        c3 = __builtin_amdgcn_wmma_f32_16x16x4_f32(false, a, false, b3, (short)0, c3, false, false);
    }
}

__global__ __launch_bounds__(128)
void conv_gemm_kernel(const float* __restrict__ Xd, const float* __restrict__ Ag,
                      const float* __restrict__ A, const float* __restrict__ B,
                      const float* __restrict__ bias, float* __restrict__ out,
                      int N, int K, int combine) {
    const int lane = threadIdx.x & 31;
    const int wave = threadIdx.x >> 5;
    const int row0 = (blockIdx.x * 4 + wave) * 16;
    if (row0 >= N) return;  // wave-uniform
    const int m = lane & 15;
    const int khi = lane >> 4;
    int rowA = row0 + m;
    if (rowA > N - 1) rowA = N - 1;  // clamp loads; stores guarded below

    v8f c0 = {}, c1 = {}, c2 = {}, c3 = {};
    wmma_accum(Xd, A, rowA, K, m, khi, c0, c1, c2, c3);
    wmma_accum(Ag, B, rowA, K, m, khi, c0, c1, c2, c3);

#pragma unroll
    for (int r = 0; r < 8; ++r) {
        int row = row0 + r + 8 * khi;  // C layout: VGPR r -> M=r (lanes 0-15) / M=r+8
        if (row < N) {
            float* op = out + (size_t)row * 64 + m;
            float v0 = c0[r] + bias[m];
            float v1 = c1[r] + bias[m + 16];
            float v2 = c2[r] + bias[m + 32];
            float v3 = c3[r] + bias[m + 48];
            if (combine) {
                v0 = (op[0]  + v0) * 0.5f;
                v1 = (op[16] + v1) * 0.5f;
                v2 = (op[32] + v2) * 0.5f;
                v3 = (op[48] + v3) * 0.5f;
            }
            op[0] = v0; op[16] = v1; op[32] = v2; op[48] = v3;
        }
    }
}

// ---------------- BatchNorm ----------------------------------
__global__ __launch_bounds__(256)
void bn_stats_kernel(const float* __restrict__ X, float* __restrict__ sums, int N) {
    __shared__ float ss[128];
    int t = threadIdx.x;
    if (t < 128) ss[t] = 0.f;
    __syncthreads();
    int col = t & 63;
    int rl = t >> 6;  // 0..3
    int base = blockIdx.x * 64;
    float s = 0.f, sq = 0.f;
    for (int i = rl; i < 64; i += 4) {
        int row = base + i;
        if (row < N) {
            float v = X[(size_t)row * 64 + col];
            s += v; sq += v * v;
        }
    }
    atomicAdd(&ss[col], s);
    atomicAdd(&ss[64 + col], sq);
    __syncthreads();
    if (t < 64) {
        atomicAdd(&sums[t], ss[t]);
        atomicAdd(&sums[64 + t], ss[64 + t]);
    }
}

__global__ void bn_finalize_kernel(const float* __restrict__ sums, const float* __restrict__ g,
                                   const float* __restrict__ b, float Ninv,
                                   float* __restrict__ ss) {
    int j = threadIdx.x;
    if (j >= 64) return;
    float mean = sums[j] * Ninv;
    float var = sums[64 + j] * Ninv - mean * mean;
    float sc = g[j] * rsqrtf(var + BN_EPS);
    ss[j] = sc;
    ss[64 + j] = b[j] - mean * sc;
}

__global__ void bn_apply_kernel(float* __restrict__ X, const float* __restrict__ ss,
                                long long total, int leaky) {
    long long i = blockIdx.x * (long long)blockDim.x + threadIdx.x;
    if (i >= total) return;
    int col = (int)(i & 63);
    float v = X[i] * ss[col] + ss[64 + col];
    if (leaky) v = v > 0.f ? v : SLOPE * v;
    X[i] = v;
}

// ---------------- label-edge dot products --------------------
__global__ void dot_kernel(const float* __restrict__ Xs, const float* __restrict__ Xt,
                           const int* __restrict__ lbl, float* __restrict__ out, int E) {
    int i = blockIdx.x * blockDim.x + threadIdx.x;
    if (i >= E) return;
    int a = lbl[i], b = lbl[E + i];
    const float4* pa = (const float4*)(Xs + (size_t)a * 64);
    const float4* pb = (const float4*)(Xt + (size_t)b * 64);
    float s = 0.f;
#pragma unroll
    for (int k = 0; k < 16; ++k) {
        float4 va = pa[k], vb = pb[k];
        s += va.x * vb.x + va.y * vb.y + va.z * vb.z + va.w * vb.w;
    }
    out[i] = s;
}

// ---------------- host orchestration -------------------------
struct ConvP { const float *Ws, *bs, *Wd, *bd, *Wu, *bu; };

extern "C" void kernel_launch(void* const* d_in, const int* in_sizes, int n_in,
                              void* d_out, int out_size, void* d_ws, size_t ws_size,
                              hipStream_t stream) {
    (void)n_in; (void)out_size; (void)ws_size;
    auto Fp = [&](int i) { return (const float*)d_in[i]; };
    auto Ip = [&](int i) { return (const int*)d_in[i]; };

    const float *x_gene, *x_disease;
    const int *ei_g2d, *ei_d2g, *ei_ppi, *lbl_g2d, *lbl_d2g, *lbl_ppi;
    ConvP cv[6];  // 0:l1.g2d 1:l1.d2g 2:l1.ppi 3:l2.g2d 4:l2.d2g 5:l2.ppi
    const float *bn1g_g, *bn1g_b, *bn1d_g, *bn1d_b, *bn2g_g, *bn2g_b, *bn2d_g, *bn2d_b;

    bool insertion = (in_sizes[0] == N_GENE * F_IN);
    if (insertion) {  // dict insertion order, nested dicts depth-first
        x_gene = Fp(0); x_disease = Fp(1);
        auto mk = [&](int b) { return ConvP{Fp(b), Fp(b+1), Fp(b+2), Fp(b+3), Fp(b+4), Fp(b+5)}; };
        cv[0] = mk(2);  cv[1] = mk(8);  cv[2] = mk(14);
        cv[3] = mk(20); cv[4] = mk(26); cv[5] = mk(32);
        bn1g_g = Fp(38); bn1g_b = Fp(39); bn1d_g = Fp(40); bn1d_b = Fp(41);
        bn2g_g = Fp(42); bn2g_b = Fp(43); bn2d_g = Fp(44); bn2d_b = Fp(45);
        ei_g2d = Ip(46); ei_d2g = Ip(47); ei_ppi = Ip(48);
        lbl_g2d = Ip(49); lbl_d2g = Ip(50); lbl_ppi = Ip(51);
    } else {          // jax tree_flatten (sorted-key) order
        ei_d2g = Ip(0); ei_g2d = Ip(1); ei_ppi = Ip(2);
        lbl_d2g = Ip(3); lbl_g2d = Ip(4); lbl_ppi = Ip(5);
        bn1d_b = Fp(6); bn1d_g = Fp(7); bn1g_b = Fp(8); bn1g_g = Fp(9);
        bn2d_b = Fp(10); bn2d_g = Fp(11); bn2g_b = Fp(12); bn2g_g = Fp(13);
        auto mk = [&](int b) { ConvP c; c.Wd = Fp(b); c.Ws = Fp(b+1); c.Wu = Fp(b+2);
                               c.bd = Fp(b+3); c.bs = Fp(b+4); c.bu = Fp(b+5); return c; };
        cv[1] = mk(14); cv[0] = mk(20); cv[2] = mk(26);   // l1: d2g,g2d,ppi
        cv[4] = mk(32); cv[3] = mk(38); cv[5] = mk(44);   // l2: d2g,g2d,ppi
        x_disease = Fp(50); x_gene = Fp(51);
    }
    float* out = (float*)d_out;

    // workspace bump allocator
    size_t off = 0;
    auto alloc = [&](size_t bytes) {
        void* p = (char*)d_ws + off;
        off += (bytes + 255) & ~(size_t)255;
        return p;
    };
    float* aggr_d  = (float*)alloc((size_t)N_DIS  * F_IN * 4);
    float* aggr_g1 = (float*)alloc((size_t)N_GENE * F_IN * 4);
    float* aggr_g2 = (float*)alloc((size_t)N_GENE * F_IN * 4);
    float* cnt_d   = (float*)alloc((size_t)N_DIS  * 4);
    float* cnt_g1  = (float*)alloc((size_t)N_GENE * 4);
    float* cnt_g2  = (float*)alloc((size_t)N_GENE * 4);
    float* xg1 = (float*)alloc((size_t)N_GENE * H_DIM * 4);
    float* xd1 = (float*)alloc((size_t)N_DIS  * H_DIM * 4);
    float* xg2 = (float*)alloc((size_t)N_GENE * H_DIM * 4);
    float* xd2 = (float*)alloc((size_t)N_DIS  * H_DIM * 4);
    float* fw  = (float*)alloc((size_t)6 * (2 * 128 * 64 + 64) * 4);
    float* bnsum = (float*)alloc(128 * 4);
    float* bnss  = (float*)alloc(128 * 4);
    auto fwA = [&](int i) { return fw + (size_t)i * 16448; };
    auto fwB = [&](int i) { return fw + (size_t)i * 16448 + 8192; };
    auto fwc = [&](int i) { return fw + (size_t)i * 16448 + 16384; };

    auto blocks = [](long long total, int bs) { return (unsigned)((total + bs - 1) / bs); };

    // 0) fuse weights (6 convs)
    for (int i = 0; i < 6; ++i) {
        int F = (i < 3) ? 128 : 64;
        fuse_weights_kernel<<<blocks(128 * 64, 256), 256, 0, stream>>>(
            cv[i].Ws, cv[i].bs, cv[i].Wd, cv[i].bd, cv[i].Wu, cv[i].bu,
            fwA(i), fwB(i), fwc(i), F, F);
    }

    // ================= layer 1 (F=128) =================
    hipMemsetAsync(aggr_d,  0, (size_t)N_DIS  * F_IN * 4, stream);
    hipMemsetAsync(aggr_g1, 0, (size_t)N_GENE * F_IN * 4, stream);
    hipMemsetAsync(aggr_g2, 0, (size_t)N_GENE * F_IN * 4, stream);
    hipMemsetAsync(cnt_d,  0, (size_t)N_DIS  * 4, stream);
    hipMemsetAsync(cnt_g1, 0, (size_t)N_GENE * 4, stream);
    hipMemsetAsync(cnt_g2, 0, (size_t)N_GENE * 4, stream);

    scatter_add_kernel<<<blocks((long long)E_GDA * 32, 256), 256, 0, stream>>>(
        x_gene, ei_g2d, aggr_d, cnt_d, E_GDA, F_IN);
    scatter_add_kernel<<<blocks((long long)E_GDA * 32, 256), 256, 0, stream>>>(
        x_disease, ei_d2g, aggr_g1, cnt_g1, E_GDA, F_IN);
    scatter_add_kernel<<<blocks((long long)E_PPI * 32, 256), 256, 0, stream>>>(
        x_gene, ei_ppi, aggr_g2, cnt_g2, E_PPI, F_IN);

    mean_div_kernel<<<blocks((long long)N_DIS * F_IN, 256), 256, 0, stream>>>(
        aggr_d, cnt_d, (long long)N_DIS * F_IN, 7);
    mean_div_kernel<<<blocks((long long)N_GENE * F_IN, 256), 256, 0, stream>>>(
        aggr_g1, cnt_g1, (long long)N_GENE * F_IN, 7);
    mean_div_kernel<<<blocks((long long)N_GENE * F_IN, 256), 256, 0, stream>>>(
        aggr_g2, cnt_g2, (long long)N_GENE * F_IN, 7);

    conv_gemm_kernel<<<blocks(N_DIS, 64), 128, 0, stream>>>(
        x_disease, aggr_d, fwA(0), fwB(0), fwc(0), xd1, N_DIS, 128, 0);
    conv_gemm_kernel<<<blocks(N_GENE, 64), 128, 0, stream>>>(
        x_gene, aggr_g1, fwA(1), fwB(1), fwc(1), xg1, N_GENE, 128, 0);
    conv_gemm_kernel<<<blocks(N_GENE, 64), 128, 0, stream>>>(
        x_gene, aggr_g2, fwA(2), fwB(2), fwc(2), xg1, N_GENE, 128, 1);  // (g1+g2)*0.5

    // BN1 gene (+leaky)
    hipMemsetAsync(bnsum, 0, 128 * 4, stream);
    bn_stats_kernel<<<blocks(N_GENE, 64), 256, 0, stream>>>(xg1, bnsum, N_GENE);
    bn_finalize_kernel<<<1, 64, 0, stream>>>(bnsum, bn1g_g, bn1g_b, 1.0f / N_GENE, bnss);
    bn_apply_kernel<<<blocks((long long)N_GENE * 64, 256), 256, 0, stream>>>(
        xg1, bnss, (long long)N_GENE * 64, 1);
    // BN1 dis (+leaky)
    hipMemsetAsync(bnsum, 0, 128 * 4, stream);
    bn_stats_kernel<<<blocks(N_DIS, 64), 256, 0, stream>>>(xd1, bnsum, N_DIS);
    bn_finalize_kernel<<<1, 64, 0, stream>>>(bnsum, bn1d_g, bn1d_b, 1.0f / N_DIS, bnss);
    bn_apply_kernel<<<blocks((long long)N_DIS * 64, 256), 256, 0, stream>>>(
        xd1, bnss, (long long)N_DIS * 64, 1);

    // ================= layer 2 (F=64) =================
    hipMemsetAsync(aggr_d,  0, (size_t)N_DIS  * H_DIM * 4, stream);
    hipMemsetAsync(aggr_g1, 0, (size_t)N_GENE * H_DIM * 4, stream);
    hipMemsetAsync(aggr_g2, 0, (size_t)N_GENE * H_DIM * 4, stream);
    hipMemsetAsync(cnt_d,  0, (size_t)N_DIS  * 4, stream);
    hipMemsetAsync(cnt_g1, 0, (size_t)N_GENE * 4, stream);
    hipMemsetAsync(cnt_g2, 0, (size_t)N_GENE * 4, stream);

    scatter_add_kernel<<<blocks((long long)E_GDA * 16, 256), 256, 0, stream>>>(
        xg1, ei_g2d, aggr_d, cnt_d, E_GDA, H_DIM);
    scatter_add_kernel<<<blocks((long long)E_GDA * 16, 256), 256, 0, stream>>>(
        xd1, ei_d2g, aggr_g1, cnt_g1, E_GDA, H_DIM);
    scatter_add_kernel<<<blocks((long long)E_PPI * 16, 256), 256, 0, stream>>>(
        xg1, ei_ppi, aggr_g2, cnt_g2, E_PPI, H_DIM);

    mean_div_kernel<<<blocks((long long)N_DIS * H_DIM, 256), 256, 0, stream>>>(
        aggr_d, cnt_d, (long long)N_DIS * H_DIM, 6);
    mean_div_kernel<<<blocks((long long)N_GENE * H_DIM, 256), 256, 0, stream>>>(
        aggr_g1, cnt_g1, (long long)N_GENE * H_DIM, 6);
    mean_div_kernel<<<blocks((long long)N_GENE * H_DIM, 256), 256, 0, stream>>>(
        aggr_g2, cnt_g2, (long long)N_GENE * H_DIM, 6);

    conv_gemm_kernel<<<blocks(N_DIS, 64), 128, 0, stream>>>(
        xd1, aggr_d, fwA(3), fwB(3), fwc(3), xd2, N_DIS, 64, 0);
    conv_gemm_kernel<<<blocks(N_GENE, 64), 128, 0, stream>>>(
        xg1, aggr_g1, fwA(4), fwB(4), fwc(4), xg2, N_GENE, 64, 0);
    conv_gemm_kernel<<<blocks(N_GENE, 64), 128, 0, stream>>>(
        xg1, aggr_g2, fwA(5), fwB(5), fwc(5), xg2, N_GENE, 64, 1);

    // BN2 gene / dis (no leaky)
    hipMemsetAsync(bnsum, 0, 128 * 4, stream);
    bn_stats_kernel<<<blocks(N_GENE, 64), 256, 0, stream>>>(xg2, bnsum, N_GENE);
    bn_finalize_kernel<<<1, 64, 0, stream>>>(bnsum, bn2g_g, bn2g_b, 1.0f / N_GENE, bnss);
    bn_apply_kernel<<<blocks((long long)N_GENE * 64, 256), 256, 0, stream>>>(
        xg2, bnss, (long long)N_GENE * 64, 0);
    hipMemsetAsync(bnsum, 0, 128 * 4, stream);
    bn_stats_kernel<<<blocks(N_DIS, 64), 256, 0, stream>>>(xd2, bnsum, N_DIS);
    bn_finalize_kernel<<<1, 64, 0, stream>>>(bnsum, bn2d_g, bn2d_b, 1.0f / N_DIS, bnss);
    bn_apply_kernel<<<blocks((long long)N_DIS * 64, 256), 256, 0, stream>>>(
        xd2, bnss, (long long)N_DIS * 64, 0);

    // ================= label dot products =================
    dot_kernel<<<blocks(E_LBL, 256), 256, 0, stream>>>(xg2, xd2, lbl_g2d, out, E_LBL);
    dot_kernel<<<blocks(E_LBL, 256), 256, 0, stream>>>(xd2, xg2, lbl_d2g, out + E_LBL, E_LBL);
    dot_kernel<<<blocks(E_LBL, 256), 256, 0, stream>>>(xg2, xg2, lbl_ppi, out + 2 * E_LBL, E_LBL);
}